// LinearEncoder_41291815584492
// MI455X (gfx1250) — compile-verified
//
#include <hip/hip_runtime.h>
#include <hip/hip_bf16.h>

// GCNConv forward for MI455X (gfx1250, wave32).
//   x:[N,128] f32, edge_index:[2,E] int, W:[128,256] f32, b:[256] f32
//   out:[N,256] f32
// Pipeline: degree -> dinv -> h = x@W (V_WMMA_F32_16X16X4_F32) ->
//           out = b + h*dinv^2 (self loop) -> edge scatter via f32 atomics.
//
// Workspace layout in d_ws:
//   [0              .. 64KB)  : dinv  (N floats)
//   [64KB           .. 128KB) : deg   (N u32)
//   [128KB          .. +N*256*4) : h  (N x 256 floats, ~10.2 MB; L2-resident)

#define KDIM 128
#define ODIM 256

typedef __attribute__((ext_vector_type(2))) float v2f;
typedef __attribute__((ext_vector_type(8))) float v8f;

__global__ void gcn_deg_init(unsigned* __restrict__ deg, int n) {
    int i = blockIdx.x * blockDim.x + threadIdx.x;
    if (i < n) deg[i] = 1u;  // self-loop contributes 1 to every node's degree
}

__global__ void gcn_deg_count(const int* __restrict__ ei, unsigned* __restrict__ deg,
                              int E) {
    int e = blockIdx.x * blockDim.x + threadIdx.x;
    if (e < E) {
        int c = ei[E + e];  // target node (col)
        atomicAdd(&deg[c], 1u);
    }
}

__global__ void gcn_dinv(const unsigned* __restrict__ deg, float* __restrict__ dinv,
                         int n) {
    int i = blockIdx.x * blockDim.x + threadIdx.x;
    if (i < n) {
        float d = (float)deg[i];
        dinv[i] = (d > 0.0f) ? rsqrtf(d) : 0.0f;
    }
}

// h = x @ W  via V_WMMA_F32_16X16X4_F32.
// One wave -> one 16x16 tile of h. grid = (M/16, ODIM/16/4), block = 128 (4 waves).
// A layout (ISA 7.12.2, 32-bit A 16x4): lanes 0-15 hold K=k+0 (v0), K=k+1 (v1)
// for row M=lane; lanes 16-31 hold K=k+2, K=k+3 for row M=lane-16.
// B mirrored: lanes 0-15 hold rows K=k+0/k+1 at col N=lane; lanes 16-31 K=k+2/k+3.
// C/D: VGPR r holds (M = r + 8*(lane>=16), N = lane%16).
__global__ void gcn_gemm_wmma(const float* __restrict__ x, const float* __restrict__ W,
                              float* __restrict__ h) {
    const int lane = threadIdx.x & 31;
    const int wave = threadIdx.x >> 5;               // 0..3
    const int m0 = blockIdx.x * 16;                  // 10000 rows = 625 tiles exactly
    const int n0 = (blockIdx.y * 4 + wave) * 16;     // 16 N-tiles
    const int half = lane >> 4;                      // 0 or 1
    const int l16  = lane & 15;
    const int am = m0 + l16;                         // A row for this lane
    const int bn = n0 + l16;                         // B col for this lane

    v8f c = {};
    #pragma unroll 4
    for (int k = 0; k < KDIM; k += 4) {
        const int ka = k + half * 2;
        // A: contiguous float2 load from x[am][ka..ka+1]
        v2f a = *(const v2f*)(x + (size_t)am * KDIM + ka);
        // B: rows ka and ka+1 of W at column bn
        v2f b;
        b.x = W[(size_t)ka * ODIM + bn];
        b.y = W[(size_t)(ka + 1) * ODIM + bn];
        c = __builtin_amdgcn_wmma_f32_16x16x4_f32(
                /*neg_a=*/false, a, /*neg_b=*/false, b,
                /*c_mod=*/(short)0, c, /*reuse_a=*/false, /*reuse_b=*/false);
    }
    #pragma unroll
    for (int r = 0; r < 8; ++r) {
        h[(size_t)(m0 + r + half * 8) * ODIM + (n0 + l16)] = c[r];
    }
}

// out[i][j] = b[j] + h[i][j] * dinv[i]^2   (bias + self-loop message)
__global__ void gcn_init_out(const float* __restrict__ h, const float* __restrict__ dinv,
                             const float* __restrict__ b, float* __restrict__ out,
                             int total) {
    int t = blockIdx.x * blockDim.x + threadIdx.x;
    if (t < total) {
        int i = t >> 8;       // node
        int j = t & 255;      // feature
        float di = dinv[i];
        out[t] = b[j] + h[t] * di * di;
    }
}

// Edge scatter: one wave per edge; lane owns 8 contiguous features (two float4
// loads of h[row], fully coalesced 1KB per wave), then 8 scalar f32 atomics
// into out[col]. h and out are L2-resident (10.2 MB each vs 192 MB L2).
__global__ void gcn_scatter(const int* __restrict__ ei, const float* __restrict__ h,
                            const float* __restrict__ dinv, float* __restrict__ out,
                            int E) {
    long long gid = (long long)blockIdx.x * blockDim.x + threadIdx.x;
    int e    = (int)(gid >> 5);
    int lane = (int)(gid & 31);
    if (e >= E) return;

    int r = ei[e];        // source (row)
    int c = ei[E + e];    // target (col)
    float norm = dinv[r] * dinv[c];

    const float4* hv = (const float4*)(h + (size_t)r * ODIM);
    float4 v0 = hv[lane * 2 + 0];
    float4 v1 = hv[lane * 2 + 1];

    float* o = out + (size_t)c * ODIM + lane * 8;
    atomicAdd(o + 0, v0.x * norm);
    atomicAdd(o + 1, v0.y * norm);
    atomicAdd(o + 2, v0.z * norm);
    atomicAdd(o + 3, v0.w * norm);
    atomicAdd(o + 4, v1.x * norm);
    atomicAdd(o + 5, v1.y * norm);
    atomicAdd(o + 6, v1.z * norm);
    atomicAdd(o + 7, v1.w * norm);
}

extern "C" void kernel_launch(void* const* d_in, const int* in_sizes, int n_in,
                              void* d_out, int out_size, void* d_ws, size_t ws_size,
                              hipStream_t stream) {
    const float* x  = (const float*)d_in[0];   // [N,128]
    const int*   ei = (const int*)d_in[1];     // [2,E]
    const float* W  = (const float*)d_in[2];   // [128,256]
    const float* b  = (const float*)d_in[3];   // [256]
    float* out = (float*)d_out;                // [N,256]

    const int N = in_sizes[0] / KDIM;          // 10000
    const int E = in_sizes[1] / 2;             // 640000

    float*    dinv = (float*)d_ws;
    unsigned* deg  = (unsigned*)((char*)d_ws + (64u << 10));
    float*    h    = (float*)((char*)d_ws + (128u << 10));

    // 1) degree (init to 1 for self-loops, then count targets)
    gcn_deg_init<<<(N + 255) / 256, 256, 0, stream>>>(deg, N);
    gcn_deg_count<<<(E + 255) / 256, 256, 0, stream>>>(ei, deg, E);

    // 2) dinv = rsqrt(deg)
    gcn_dinv<<<(N + 255) / 256, 256, 0, stream>>>(deg, dinv, N);

    // 3) h = x @ W via f32 WMMA (exact M tiling: 10000 = 625 * 16)
    dim3 ggrid(N / 16, ODIM / 16 / 4);
    gcn_gemm_wmma<<<ggrid, 128, 0, stream>>>(x, W, h);

    // 4) out = b + self-loop term (also overwrites poisoned d_out)
    int total = N * ODIM;
    gcn_init_out<<<(total + 255) / 256, 256, 0, stream>>>(h, dinv, b, out, total);

    // 5) per-edge scatter: one wave per edge
    long long threads = (long long)E * 32;
    int blocks = (int)((threads + 255) / 256);
    gcn_scatter<<<blocks, 256, 0, stream>>>(ei, h, dinv, out, E);
}